// CropSampler_59158879535309
// MI455X (gfx1250) — compile-verified
//
#include <hip/hip_runtime.h>

typedef __attribute__((ext_vector_type(2))) float v2f;
typedef __attribute__((ext_vector_type(4))) float f4v;
typedef __attribute__((ext_vector_type(8))) float v8f;

#define BF      256      // B*C flat batch
#define N_PTS   1024     // centers per batch elem
#define D_DIM   3
#define K_CROP  512
#define ROW_F   96       // G*D floats per patch row
#define ROW_F4  24       // float4 per patch row

// One workgroup per flat batch element. 256 threads = 8 waves.
// Phase 1: d2 via V_WMMA_F32_16X16X4_F32 (A row = (-2q, 1), B col = (c, |c|^2+|q|^2)).
// Phase 2: bitonic sort of packed (f32bits(d2) << 32 | idx) in LDS, ascending.
// Phase 3: coalesced b128 gather of 512 selected rows (nontemporal stores).
__global__ __launch_bounds__(256) void crop_sampler_kernel(
    const float* __restrict__ patches,
    const float* __restrict__ centers,
    const int*   __restrict__ center_idx,
    float* __restrict__ out_patches,
    float* __restrict__ out_centers)
{
    __shared__ unsigned long long keys[N_PTS];

    const int bf   = blockIdx.x;
    const int tid  = threadIdx.x;
    const int wave = tid >> 5;
    const int lane = tid & 31;

    const float* cbase = centers + (size_t)bf * N_PTS * D_DIM;

    // Query center (uniform across the block).
    const int   qi = center_idx[bf];
    const float q0 = cbase[qi * 3 + 0];
    const float q1 = cbase[qi * 3 + 1];
    const float q2 = cbase[qi * 3 + 2];
    const float qq = q0 * q0 + q1 * q1 + q2 * q2;

    const bool hi = (lane >= 16);   // ISA A layout: lanes 0-15 hold K=0,1; 16-31 hold K=2,3
    v2f a;
    a.x = hi ? (-2.0f * q2) : (-2.0f * q0);
    a.y = hi ? 1.0f         : (-2.0f * q1);

    // ---- Phase 1: distances via WMMA (16 centers per wave per iteration) ----
    for (int it = 0; it < 8; ++it) {
        const int n  = it * 128 + wave * 16 + (lane & 15);
        const float cx = cbase[n * 3 + 0];
        const float cy = cbase[n * 3 + 1];
        const float cz = cbase[n * 3 + 2];
        v2f b;
        b.x = hi ? cz : cx;
        b.y = hi ? (cx * cx + cy * cy + cz * cz + qq) : cy;
        v8f acc = {};
        acc = __builtin_amdgcn_wmma_f32_16x16x4_f32(
            /*neg_a=*/false, a, /*neg_b=*/false, b,
            /*c_mod=*/(short)0, acc, /*reuse_a=*/false, /*reuse_b=*/false);
        float d2 = acc[0];                 // all rows identical; N = lane & 15
        d2 = d2 < 0.0f ? 0.0f : d2;        // guard FP cancellation near q==c
        if (lane < 16) {
            unsigned fb = __float_as_uint(d2);
            keys[n] = ((unsigned long long)fb << 32) | (unsigned)n;
        }
    }
    __syncthreads();

    // ---- Phase 2: bitonic sort, ascending (ties -> smaller index first) ----
    for (unsigned k = 2; k <= N_PTS; k <<= 1) {
        for (unsigned j = k >> 1; j > 0; j >>= 1) {
            for (unsigned i = tid; i < N_PTS; i += 256) {
                unsigned l = i ^ j;
                if (l > i) {
                    unsigned long long va = keys[i];
                    unsigned long long vb = keys[l];
                    bool up = ((i & k) == 0);
                    if ((va > vb) == up) { keys[i] = vb; keys[l] = va; }
                }
            }
            __syncthreads();
        }
    }

    // ---- Phase 3a: gather cropped centers (512 x 3 floats) ----
    float* ocent = out_centers + (size_t)bf * K_CROP * D_DIM;
    for (int o = tid; o < K_CROP * D_DIM; o += 256) {
        int r = o / 3;
        int d = o - r * 3;
        int src = (int)(keys[r] & 1023u);
        ocent[o] = cbase[src * 3 + d];
    }

    // ---- Phase 3b: gather cropped patches (512 rows x 24 float4) ----
    const f4v* p4 = (const f4v*)(patches + (size_t)bf * N_PTS * ROW_F);
    f4v*       o4 = (f4v*)(out_patches + (size_t)bf * K_CROP * ROW_F);
    for (int o = tid; o < K_CROP * ROW_F4; o += 256) {
        int r  = o / ROW_F4;
        int c4 = o - r * ROW_F4;
        int src = (int)(keys[r] & 1023u);
        f4v v = p4[(size_t)src * ROW_F4 + c4];
        __builtin_nontemporal_store(v, &o4[o]);   // single-use output: bypass cache retention
    }
}

extern "C" void kernel_launch(void* const* d_in, const int* in_sizes, int n_in,
                              void* d_out, int out_size, void* d_ws, size_t ws_size,
                              hipStream_t stream) {
    const float* patches    = (const float*)d_in[0];
    const float* centers    = (const float*)d_in[1];
    const int*   center_idx = (const int*)d_in[2];
    // d_out = cropped_patches (B,C,K,G,D) flat, then cropped_centers (B,C,K,D) flat
    float* out_patches = (float*)d_out;
    float* out_centers = out_patches + (size_t)BF * K_CROP * ROW_F;   // 12,582,912

    crop_sampler_kernel<<<dim3(BF), dim3(256), 0, stream>>>(
        patches, centers, center_idx, out_patches, out_centers);

    (void)in_sizes; (void)n_in; (void)out_size; (void)d_ws; (void)ws_size;
}